// EmbeddingModel_10419590660204
// MI455X (gfx1250) — compile-verified
//
#include <hip/hip_runtime.h>
#include <hip/hip_bf16.h>

// ---------------------------------------------------------------------------
// 2-layer LSTM + head for MI455X (gfx1250, wave32, WMMA).
//
//   K0 zero_ws     : zero c-state, h0, hbuf, sync counter
//   K1 gemm_g1     : G1[8192,2048] = x_flat @ W_ih1^T + b_ih1 + b_hh1
//                    (f32 WMMA, 32x64 per wave: 8 acc, 0.75 loads/WMMA)
//   K2 lstm1_step  : x128 launches; gates = h_prev @ W_hh1^T + G1[t];
//                    4 waves = 4 gates of one 16x16 tile, LDS exchange + update
//   K3 gemm_g2     : G2[8192,1024] = hs1 @ W_ih2^T + b_ih2 + b_hh2 (as K1)
//   K4 lstm2_seq   : 16 persistent WGs, 8192 serial batch-1 steps;
//                    W_hh2 staged to LDS via GLOBAL_LOAD_ASYNC_TO_LDS_B128,
//                    global release/acquire counter sync, double-buffered h
//   K5 head        : emb = h+h^2 ; @W1^T+b1 ; @W2^T+b2 -> out[10]
// ---------------------------------------------------------------------------

typedef float v2f __attribute__((ext_vector_type(2)));
typedef float v8f __attribute__((ext_vector_type(8)));

#define B_     64
#define T_     128
#define D_     768
#define H1_    512
#define H2_    256
#define G1COLS 2048   // 4*H1
#define G2COLS 1024   // 4*H2
#define S_     (T_ * B_)  // 8192

__device__ __forceinline__ float sigf(float x) { return 1.0f / (1.0f + __expf(-x)); }

// ---------------------------------------------------------------------------
// K0: zero the stateful region (runs every call; graph replays don't re-init).
// ---------------------------------------------------------------------------
__global__ void zero_ws(float* __restrict__ p, int n) {
    int i = blockIdx.x * blockDim.x + threadIdx.x;
    if (i < n) p[i] = 0.0f;
}

// ---------------------------------------------------------------------------
// Register-blocked f32-WMMA GEMM body: wave computes a 32x64 tile of
// out = A @ W^T + bias, K = kdim.  A rows are fetched through arow0/arow1
// (per-lane row pointers), W rows through 4 per-lane col pointers.
// WMMA f32 16x16x4 lane layout:
//   A (16x4): lane l -> M = l&15, K = (l>>4)*2 + {v0,v1}
//   B (4x16): lane l -> N = l&15, K = (l>>4)*2 + {v0,v1}
//   C (16x16): vgpr r, lane l -> row = r + (l>>4)*8, col = l&15
// ---------------------------------------------------------------------------
__device__ __forceinline__ void gemm_tile_32x64(
    const float* arow0, const float* arow1,
    const float* brow0, const float* brow1,
    const float* brow2, const float* brow3,
    int kdim, v8f acc[2][4]) {
    for (int k0 = 0; k0 < kdim; k0 += 4) {
        v2f a0 = *(const v2f*)(arow0 + k0);
        v2f a1 = *(const v2f*)(arow1 + k0);
        v2f b0 = *(const v2f*)(brow0 + k0);
        v2f b1 = *(const v2f*)(brow1 + k0);
        v2f b2 = *(const v2f*)(brow2 + k0);
        v2f b3 = *(const v2f*)(brow3 + k0);
        acc[0][0] = __builtin_amdgcn_wmma_f32_16x16x4_f32(false, a0, false, b0, (short)0, acc[0][0], false, false);
        acc[0][1] = __builtin_amdgcn_wmma_f32_16x16x4_f32(false, a0, false, b1, (short)0, acc[0][1], false, false);
        acc[0][2] = __builtin_amdgcn_wmma_f32_16x16x4_f32(false, a0, false, b2, (short)0, acc[0][2], false, false);
        acc[0][3] = __builtin_amdgcn_wmma_f32_16x16x4_f32(false, a0, false, b3, (short)0, acc[0][3], false, false);
        acc[1][0] = __builtin_amdgcn_wmma_f32_16x16x4_f32(false, a1, false, b0, (short)0, acc[1][0], false, false);
        acc[1][1] = __builtin_amdgcn_wmma_f32_16x16x4_f32(false, a1, false, b1, (short)0, acc[1][1], false, false);
        acc[1][2] = __builtin_amdgcn_wmma_f32_16x16x4_f32(false, a1, false, b2, (short)0, acc[1][2], false, false);
        acc[1][3] = __builtin_amdgcn_wmma_f32_16x16x4_f32(false, a1, false, b3, (short)0, acc[1][3], false, false);
    }
}

// ---------------------------------------------------------------------------
// K1: G1 = x_flat @ W_ih1^T + (b_ih1 + b_hh1).
// x_flat row s = t*B + b maps to x[b, t, :].
// Block = 128 (4 waves); wave w owns N-range [blockIdx.y*256 + w*64, +64).
// Grid = (8192/32, 2048/256) = (256, 8).
// ---------------------------------------------------------------------------
__global__ void gemm_g1(const float* __restrict__ x,
                        const float* __restrict__ Wih1,
                        const float* __restrict__ bih1,
                        const float* __restrict__ bhh1,
                        float* __restrict__ G1) {
    const int lane = threadIdx.x & 31;
    const int wave = threadIdx.x >> 5;
    const int m0   = blockIdx.x * 32;
    const int n0   = blockIdx.y * 256 + wave * 64;
    const int lr   = lane & 15;
    const int kp   = (lane >> 4) * 2;

    const int s0 = m0 + lr, s1 = m0 + 16 + lr;
    const float* arow0 = x + (size_t)(s0 & 63) * (T_ * D_) + (size_t)(s0 >> 6) * D_ + kp;
    const float* arow1 = x + (size_t)(s1 & 63) * (T_ * D_) + (size_t)(s1 >> 6) * D_ + kp;
    const float* brow0 = Wih1 + (size_t)(n0 +  0 + lr) * D_ + kp;
    const float* brow1 = Wih1 + (size_t)(n0 + 16 + lr) * D_ + kp;
    const float* brow2 = Wih1 + (size_t)(n0 + 32 + lr) * D_ + kp;
    const float* brow3 = Wih1 + (size_t)(n0 + 48 + lr) * D_ + kp;

    v8f acc[2][4] = {};
    gemm_tile_32x64(arow0, arow1, brow0, brow1, brow2, brow3, D_, acc);

    float bias[4];
#pragma unroll
    for (int nt = 0; nt < 4; ++nt) {
        int col = n0 + nt * 16 + lr;
        bias[nt] = bih1[col] + bhh1[col];
    }
#pragma unroll
    for (int mt = 0; mt < 2; ++mt)
#pragma unroll
        for (int nt = 0; nt < 4; ++nt)
#pragma unroll
            for (int r = 0; r < 8; ++r) {
                int row = m0 + mt * 16 + r + ((lane >> 4) << 3);
                int col = n0 + nt * 16 + lr;
                G1[(size_t)row * G1COLS + col] = acc[mt][nt][r] + bias[nt];
            }
}

// ---------------------------------------------------------------------------
// K2: one LSTM-1 timestep (launched 128x; stream order = recurrence).
// Wave w computes gate w's 16x16 tile of the same (batch, h) tile so the
// i/f/g/o combine is one LDS exchange.  Grid = (4, 32), wide for latency.
// ---------------------------------------------------------------------------
__global__ void lstm1_step(const float* __restrict__ hprev,  // [64,512]
                           const float* __restrict__ Whh1,   // [2048,512]
                           const float* __restrict__ G1t,    // [64,2048] slice
                           float* __restrict__ c,            // [64,512] state
                           float* __restrict__ hout) {       // hs1 + t*64*512
    __shared__ float gl[4][16][16];
    const int lane = threadIdx.x & 31;
    const int wave = threadIdx.x >> 5;
    const int m0   = blockIdx.x * 16;
    const int h0   = blockIdx.y * 16;
    const int coln = h0 + (lane & 15);
    const int kp   = (lane >> 4) * 2;

    const float* arow = hprev + (size_t)(m0 + (lane & 15)) * H1_ + kp;
    const float* brow = Whh1 + (size_t)(wave * H1_ + coln) * H1_ + kp;

    v8f acc = {};
    for (int k0 = 0; k0 < H1_; k0 += 4) {
        v2f a = *(const v2f*)(arow + k0);
        v2f b = *(const v2f*)(brow + k0);
        acc = __builtin_amdgcn_wmma_f32_16x16x4_f32(false, a, false, b,
                                                    (short)0, acc, false, false);
    }
#pragma unroll
    for (int r = 0; r < 8; ++r) {
        int row = r + ((lane >> 4) << 3);
        gl[wave][row][lane & 15] =
            acc[r] + G1t[(size_t)(m0 + row) * G1COLS + wave * H1_ + coln];
    }
    __syncthreads();

    for (int e = threadIdx.x; e < 256; e += 128) {
        int row = e >> 4, cn = e & 15;
        float ig = sigf(gl[0][row][cn]);
        float fg = sigf(gl[1][row][cn]);
        float gg = tanhf(gl[2][row][cn]);
        float og = sigf(gl[3][row][cn]);
        size_t ci = (size_t)(m0 + row) * H1_ + h0 + cn;
        float cnew = fg * c[ci] + ig * gg;
        c[ci] = cnew;
        hout[ci] = og * tanhf(cnew);
    }
}

// ---------------------------------------------------------------------------
// K3: G2 = hs1_flat @ W_ih2^T + (b_ih2 + b_hh2).  Grid = (256, 4).
// ---------------------------------------------------------------------------
__global__ void gemm_g2(const float* __restrict__ hs1,
                        const float* __restrict__ Wih2,
                        const float* __restrict__ bih2,
                        const float* __restrict__ bhh2,
                        float* __restrict__ G2) {
    const int lane = threadIdx.x & 31;
    const int wave = threadIdx.x >> 5;
    const int m0   = blockIdx.x * 32;
    const int n0   = blockIdx.y * 256 + wave * 64;
    const int lr   = lane & 15;
    const int kp   = (lane >> 4) * 2;

    const float* arow0 = hs1 + (size_t)(m0 + lr) * H1_ + kp;
    const float* arow1 = hs1 + (size_t)(m0 + 16 + lr) * H1_ + kp;
    const float* brow0 = Wih2 + (size_t)(n0 +  0 + lr) * H1_ + kp;
    const float* brow1 = Wih2 + (size_t)(n0 + 16 + lr) * H1_ + kp;
    const float* brow2 = Wih2 + (size_t)(n0 + 32 + lr) * H1_ + kp;
    const float* brow3 = Wih2 + (size_t)(n0 + 48 + lr) * H1_ + kp;

    v8f acc[2][4] = {};
    gemm_tile_32x64(arow0, arow1, brow0, brow1, brow2, brow3, H1_, acc);

    float bias[4];
#pragma unroll
    for (int nt = 0; nt < 4; ++nt) {
        int col = n0 + nt * 16 + lr;
        bias[nt] = bih2[col] + bhh2[col];
    }
#pragma unroll
    for (int mt = 0; mt < 2; ++mt)
#pragma unroll
        for (int nt = 0; nt < 4; ++nt)
#pragma unroll
            for (int r = 0; r < 8; ++r) {
                int row = m0 + mt * 16 + r + ((lane >> 4) << 3);
                int col = n0 + nt * 16 + lr;
                G2[(size_t)row * G2COLS + col] = acc[mt][nt][r] + bias[nt];
            }
}

// ---------------------------------------------------------------------------
// K4: LSTM-2, 8192 serial batch-1 steps. 16 persistent WGs x 256 threads.
// WG w owns h-dims [w*16, w*16+16) -> gate rows {d, 256+d, 512+d, 768+d}:
// 64 rows x 256 f32 = 64KB, staged into LDS (stride 257 vs bank conflicts)
// with GLOBAL_LOAD_ASYNC_TO_LDS_B128 (ASYNCcnt path, no VGPR round-trip).
// Cross-WG step sync: device-scope counter (release add / acquire spin),
// h double-buffered in global; c-state in registers of threads 0..15.
// ---------------------------------------------------------------------------
__global__ void lstm2_seq(const float* __restrict__ Whh2,  // [1024,256]
                          const float* __restrict__ G2,    // [8192,1024]
                          float* __restrict__ hbuf,        // [2][256]
                          unsigned int* __restrict__ ctr) {
    __shared__ float wlds[64 * 257];
    __shared__ float hl[H2_];
    __shared__ float gl[64];

    const int wg  = blockIdx.x;    // 0..15
    const int tid = threadIdx.x;   // 0..255

    // Async-stage this WG's 64 gate rows of Whh2 into LDS: 4096 16B chunks,
    // 16 per thread.  chunk c -> local row r = c/64, k-offset (c%64)*4.
    for (int cidx = tid; cidx < 4096; cidx += 256) {
        int r  = cidx >> 6;
        int kc = (cidx & 63) * 4;
        int gt = r >> 4, dl = r & 15;
        const float* gp = Whh2 + (size_t)(gt * H2_ + wg * 16 + dl) * H2_ + kc;
        unsigned lds_off = (unsigned)(size_t)(&wlds[r * 257 + kc]);
        unsigned long long gaddr = (unsigned long long)(size_t)gp;
        asm volatile("global_load_async_to_lds_b128 %0, %1, off"
                     :: "v"(lds_off), "v"(gaddr) : "memory");
    }
    asm volatile("s_wait_asynccnt 0x0" ::: "memory");

    const int r    = tid >> 2;                       // 0..63 local gate row
    const int p    = tid & 3;                        // K-quarter
    const int gcol = (r >> 4) * H2_ + wg * 16 + (r & 15);
    float creg = 0.0f;                               // valid for tid < 16
    __syncthreads();

    for (int s = 0; s < S_; ++s) {
        // Wait until every WG finished step s-1 (ctr == s*16).
        if (tid == 0) {
            while (__hip_atomic_load(ctr, __ATOMIC_ACQUIRE,
                                     __HIP_MEMORY_SCOPE_AGENT) < (unsigned)s * 16u) {
                __builtin_amdgcn_s_sleep(1);
            }
        }
        __syncthreads();

        hl[tid] = hbuf[(s & 1) * H2_ + tid];
        __syncthreads();

        // 64-long partial dot per thread, 4 threads per gate row.
        const float* wr = &wlds[r * 257 + p * 64];
        const float* hr = &hl[p * 64];
        float sum = 0.0f;
#pragma unroll 8
        for (int j = 0; j < 64; ++j) sum += wr[j] * hr[j];
        sum += __shfl_xor(sum, 1);
        sum += __shfl_xor(sum, 2);
        if (p == 0) {
            gl[r] = sum + G2[(size_t)s * G2COLS + gcol];
            if (s + 1 < S_)
                __builtin_prefetch(&G2[(size_t)(s + 1) * G2COLS + gcol], 0, 0);
        }
        __syncthreads();

        if (tid < 16) {
            float ig = sigf(gl[tid]);
            float fg = sigf(gl[16 + tid]);
            float gg = tanhf(gl[32 + tid]);
            float og = sigf(gl[48 + tid]);
            creg = fg * creg + ig * gg;
            hbuf[((s + 1) & 1) * H2_ + wg * 16 + tid] = og * tanhf(creg);
        }
        __syncthreads();
        if (tid == 0) {
            __hip_atomic_fetch_add(ctr, 1u, __ATOMIC_RELEASE,
                                   __HIP_MEMORY_SCOPE_AGENT);
        }
    }
}

// ---------------------------------------------------------------------------
// K5: head. hT sits in hbuf[0] (8192 steps -> final read buffer index 0).
// ---------------------------------------------------------------------------
__global__ void head(const float* __restrict__ hT,   // [256]
                     const float* __restrict__ W1, const float* __restrict__ b1,
                     const float* __restrict__ W2, const float* __restrict__ b2,
                     float* __restrict__ out) {
    __shared__ float emb[H2_];
    __shared__ float o1[128];
    int tid = threadIdx.x;  // 0..127
    for (int i = tid; i < H2_; i += 128) {
        float h = hT[i];
        emb[i] = h + h * h;
    }
    __syncthreads();
    float s = b1[tid];
    for (int k = 0; k < H2_; ++k) s += emb[k] * W1[tid * H2_ + k];
    o1[tid] = s;
    __syncthreads();
    if (tid < 10) {
        float t = b2[tid];
        for (int j = 0; j < 128; ++j) t += o1[j] * W2[tid * 128 + j];
        out[tid] = t;
    }
}

// ---------------------------------------------------------------------------
// Workspace layout (floats):
//   [0,32768) c1 | [32768,65536) h0 zeros | [65536,66048) hbuf[2][256]
//   [66048] ctr | [66560,+) G1[8192,2048], hs1[8192,512], G2[8192,1024]
// Total ~118 MB.
// ---------------------------------------------------------------------------
extern "C" void kernel_launch(void* const* d_in, const int* in_sizes, int n_in,
                              void* d_out, int out_size, void* d_ws, size_t ws_size,
                              hipStream_t stream) {
    const float* x    = (const float*)d_in[0];
    const float* Wih1 = (const float*)d_in[1];
    const float* Whh1 = (const float*)d_in[2];
    const float* bih1 = (const float*)d_in[3];
    const float* bhh1 = (const float*)d_in[4];
    const float* Wih2 = (const float*)d_in[5];
    const float* Whh2 = (const float*)d_in[6];
    const float* bih2 = (const float*)d_in[7];
    const float* bhh2 = (const float*)d_in[8];
    const float* W1   = (const float*)d_in[9];
    const float* b1   = (const float*)d_in[10];
    const float* W2   = (const float*)d_in[11];
    const float* b2   = (const float*)d_in[12];
    (void)in_sizes; (void)n_in; (void)out_size; (void)ws_size;

    float* ws   = (float*)d_ws;
    float* c1   = ws;
    float* h0z  = ws + 32768;
    float* hbuf = ws + 65536;
    unsigned int* ctr = (unsigned int*)(ws + 66048);
    float* G1  = ws + 66560;
    float* hs1 = G1 + (size_t)S_ * G1COLS;
    float* G2  = hs1 + (size_t)S_ * H1_;

    const int zn = 66560;
    zero_ws<<<(zn + 255) / 256, 256, 0, stream>>>(ws, zn);

    gemm_g1<<<dim3(S_ / 32, G1COLS / 256), 128, 0, stream>>>(x, Wih1, bih1, bhh1, G1);

    for (int t = 0; t < T_; ++t) {
        const float* hp = (t == 0) ? h0z : (hs1 + (size_t)(t - 1) * B_ * H1_);
        lstm1_step<<<dim3(B_ / 16, H1_ / 16), 128, 0, stream>>>(
            hp, Whh1, G1 + (size_t)t * B_ * G1COLS, c1,
            hs1 + (size_t)t * B_ * H1_);
    }

    gemm_g2<<<dim3(S_ / 32, G2COLS / 256), 128, 0, stream>>>(hs1, Wih2, bih2, bhh2, G2);

    lstm2_seq<<<16, 256, 0, stream>>>(Whh2, G2, hbuf, (unsigned int*)ctr);

    head<<<1, 128, 0, stream>>>(hbuf /* buffer 0 holds hT */, W1, b1, W2, b2,
                                (float*)d_out);
}